// TerpretProblem_ConcreteDistribution_13993003450850
// MI455X (gfx1250) — compile-verified
//
#include <hip/hip_runtime.h>
#include <hip/hip_bf16.h>

// CDNA5 / gfx1250 streaming Gumbel-softmax + neighbor-dot log-sum reduction.
// Memory-bound: 268 MB of `u` streamed once via async global->LDS DMA
// (global_load_async_to_lds_b64, tracked by ASYNCcnt), double-buffered so the
// async engine overlaps HBM traffic with transcendental-heavy VALU work.
// All math in hardware log2/exp2 domain (v_log_f32 / v_exp_f32 / v_rcp_f32).

#define BLK   256
#define TT    3584                 // terms per tile
#define NSLOT (TT + 1 + BLK)       // p-slots 0..TT plus per-thread dummy slots
#define NL    ((TT + 1 + BLK - 1) / BLK)   // 15 async loads per thread per tile
#define NTPB  4                    // tiles per block

// TEMPERATURE = 2.5 * 0.9999 = 2.49975
#define C1 0.57713573f             // log2(e) / TEMPERATURE
#define C2 0.40004000f             // 1 / TEMPERATURE

static __device__ __forceinline__ float2 gumbel_softmax2(float2 uv, float2 lv) {
    // z_j = (l_j + g_j)/T, g_j = -ln(-ln u_j)
    // p0 = 1/(1+exp(z1-z0));  g1-g0 = ln(ln u0 / ln u1) = ln2*log2(a0/a1),
    // a_j = log2(u_j)  (both negative -> ratio positive)
    float a0 = __builtin_amdgcn_logf(uv.x);            // v_log_f32 (log2)
    float a1 = __builtin_amdgcn_logf(uv.y);
    float r  = a0 * __builtin_amdgcn_rcpf(a1);
    float d2 = (lv.y - lv.x) * C1
             + __builtin_amdgcn_logf(r) * C2;          // log2(exp(z1-z0))
    float e  = __builtin_amdgcn_exp2f(d2);             // v_exp_f32 (exp2)
    float p0 = __builtin_amdgcn_rcpf(1.0f + e);
    float2 p; p.x = p0; p.y = 1.0f - p0;
    return p;
}

__global__ __launch_bounds__(BLK)
void terp_main_kernel(const float* __restrict__ xp,
                      const float* __restrict__ msraw,
                      const float* __restrict__ up,
                      float* __restrict__ partial,
                      int K, int bpm, int ntpb)
{
    extern __shared__ float2 smem[];             // 2 * NSLOT float2
    __shared__ float red[BLK];

    float2* sbuf0 = smem;
    float2* sbuf1 = smem + NSLOT;

    const int tid = threadIdx.x;
    const int m   = blockIdx.x / bpm;
    const int b   = blockIdx.x % bpm;
    const int t0  = b * ntpb;

    const float2* __restrict__ urow = (const float2*)up + (size_t)m * (size_t)(K - 1);
    const float2* __restrict__ msp  = (const float2*)msraw;
    const float2 xv = ((const float2*)xp)[0];

    // Issue exactly NL async b64 loads (wave-uniform count) staging p-entries
    // e = k0-1+j for slots j=0..kt into sb. Out-of-need lanes hit a private
    // dummy slot (TT+1+tid) so no LDS slot is ever written by two threads.
    auto stage = [&](float2* sb, int k0, int kt) {
        #pragma unroll
        for (int it = 0; it < NL; ++it) {
            int j = tid + it * BLK;
            int real = (j <= kt);
            int slot = real ? j : (TT + 1 + tid);
            int e = k0 - 1 + j;
            e = e < 0 ? 0 : e;
            e = e > (K - 2) ? (K - 2) : e;
            if (!real) e = 0;
            unsigned int       la = (unsigned int)(unsigned long long)(&sb[slot]);
            unsigned long long ga = (unsigned long long)(const void*)(urow + e);
            asm volatile("global_load_async_to_lds_b64 %0, %1, off"
                         :: "v"(la), "v"(ga) : "memory");
        }
    };

    float acc = 0.0f;

    int kfirst = t0 * TT;
    if (kfirst < K) {
        int cur = 0;
        {
            int kt0 = K - kfirst; if (kt0 > TT) kt0 = TT;
            stage(sbuf0, kfirst, kt0);
        }
        for (int ti = 0; ti < ntpb; ++ti) {
            int k0 = (t0 + ti) * TT;
            if (k0 >= K) break;
            int kt = K - k0; if (kt > TT) kt = TT;

            int kn = k0 + TT;
            bool hasNext = (ti + 1 < ntpb) && (kn < K);
            float2* sb  = cur ? sbuf1 : sbuf0;
            float2* sbn = cur ? sbuf0 : sbuf1;

            if (hasNext) {
                int ktn = K - kn; if (ktn > TT) ktn = TT;
                stage(sbn, kn, ktn);                      // prefetch tile ti+1
                asm volatile("s_wait_asynccnt 15" ::: "memory"); // tile ti done
            } else {
                asm volatile("s_wait_asynccnt 0" ::: "memory");  // drain
            }

            // Phase 1: p_i for i = k0..k0+kt, in place (thread reads only the
            // slots its own async loads wrote -> own asynccnt wait suffices).
            for (int j = tid; j <= kt; j += BLK) {
                int i = k0 + j;
                float2 p;
                if (i == 0 || i == K) {
                    p = xv;                                // p_0 = p_K = x
                } else {
                    float2 uv = sb[j];
                    float2 lv = msp[i - 1];                // L2-resident
                    p = gumbel_softmax2(uv, lv);
                }
                sb[j] = p;
            }
            __syncthreads();

            // Phase 2: terms k = k0+j use p[j], p[j+1]
            for (int j = tid; j < kt; j += BLK) {
                float2 a = sb[j];
                float2 c = sb[j + 1];
                float t = a.x * c.x + a.y * c.y;
                acc += __builtin_amdgcn_logf(t);
            }
            __syncthreads();    // sb may be async-overwritten next iteration
            cur ^= 1;
        }
    }

    // Deterministic block reduction
    red[tid] = acc;
    __syncthreads();
    for (int s = BLK / 2; s > 0; s >>= 1) {
        if (tid < s) red[tid] += red[tid + s];
        __syncthreads();
    }
    if (tid == 0) partial[blockIdx.x] = red[0];
}

__global__ __launch_bounds__(BLK)
void terp_reduce_kernel(const float* __restrict__ partial, int n, int M,
                        float* __restrict__ out)
{
    __shared__ double red[BLK];
    int tid = threadIdx.x;
    double s = 0.0;
    for (int i = tid; i < n; i += BLK) s += (double)partial[i];
    red[tid] = s;
    __syncthreads();
    for (int k = BLK / 2; k > 0; k >>= 1) {
        if (tid < k) red[tid] += red[tid + k];
        __syncthreads();
    }
    if (tid == 0) {
        // acc held sum of log2 -> scale by ln2; result = -sum_ln / M
        out[0] = (float)(-red[0] * 0.69314718055994530942 / (double)M);
    }
}

extern "C" void kernel_launch(void* const* d_in, const int* in_sizes, int n_in,
                              void* d_out, int out_size, void* d_ws, size_t ws_size,
                              hipStream_t stream)
{
    (void)n_in; (void)out_size; (void)ws_size;
    const float* x  = (const float*)d_in[0];
    const float* ms = (const float*)d_in[1];
    const float* u  = (const float*)d_in[2];
    float* out      = (float*)d_out;
    float* partial  = (float*)d_ws;

    const int km1x2 = in_sizes[1];          // (K-1)*2
    const int K     = km1x2 / 2 + 1;        // 1048576
    const int M     = in_sizes[2] / km1x2;  // 32

    const int ntt = (K + TT - 1) / TT;             // tiles per m row
    const int bpm = (ntt + NTPB - 1) / NTPB;       // blocks per m row
    const int nblocks = M * bpm;
    const size_t shmem = (size_t)2 * NSLOT * sizeof(float2);

    terp_main_kernel<<<nblocks, BLK, shmem, stream>>>(x, ms, u, partial,
                                                      K, bpm, NTPB);
    terp_reduce_kernel<<<1, BLK, 0, stream>>>(partial, nblocks, M, out);
}